// SIRModel_30030411333652
// MI455X (gfx1250) — compile-verified
//
#include <hip/hip_runtime.h>
#include <math.h>

// ---------------------------------------------------------------------------
// SIR graph network for MI455X (gfx1250), fp32 end-to-end.
//  - GEMMs: V_WMMA_F32_16X16X4_F32 (exact fp32; network is HBM/L2-bound so
//    fp32 matrix rate is free).
//  - Weights staged to LDS via the Tensor Data Mover (tensor_load_to_lds,
//    TENSORcnt-tracked), then repacked into a K-paired layout so each B
//    fragment is one bank-conflict-free ds_load_b64 (no VGPR repack movs).
// ---------------------------------------------------------------------------

#define DIM 128

typedef float v2f __attribute__((ext_vector_type(2)));
typedef float v8f __attribute__((ext_vector_type(8)));
typedef unsigned int u32x4 __attribute__((ext_vector_type(4)));
typedef int i32x4 __attribute__((ext_vector_type(4)));
typedef int i32x8 __attribute__((ext_vector_type(8)));

__device__ __forceinline__ float gelu_f(float x) {
    // exact (erf-based) GELU, matches jax.nn.gelu(approximate=False)
    return 0.5f * x * (1.0f + erff(x * 0.70710678118654752440f));
}

__device__ __forceinline__ void atomic_add_f32(float* p, float v) {
    __hip_atomic_fetch_add(p, v, __ATOMIC_RELAXED, __HIP_MEMORY_SCOPE_AGENT);
}

// ---------------------------------------------------------------------------
// zero-fill
// ---------------------------------------------------------------------------
__global__ void zero_f32(float* __restrict__ p, int n) {
    int i = blockIdx.x * blockDim.x + threadIdx.x;
    if (i < n) p[i] = 0.0f;
}

// ---------------------------------------------------------------------------
// in-degree via float atomics (clamped to >=1 at use site)
// ---------------------------------------------------------------------------
__global__ void deg_kernel(const int* __restrict__ dst, float* __restrict__ deg, int E) {
    int e = blockIdx.x * blockDim.x + threadIdx.x;
    if (e < E) atomic_add_f32(&deg[dst[e]], 1.0f);
}

// ---------------------------------------------------------------------------
// GEMM: C[M,128] = op(rowscale(A) @ W + bias [+ C])      (W is 128x128)
// 256 threads = 8 waves; wave w computes rows [blk*128 + w*16, +16) x all
// 128 cols as 8 WMMA tiles; K-loop = 32 x V_WMMA_F32_16X16X4_F32.
//
// W path: TDM DMA -> sLin (linear 64KB) -> repack to sWp where
//   sWp[(k>>1)*S2 + col] = (W[k&~1][col], W[(k&~1)+1][col])
// so a B fragment (K rows kr, kr+1 at 16 cols) is ONE ds_load_b64.
// S2 = 144 float2: half-wave row offset = 288 dwords = 32 banks -> the two
// 16-lane halves occupy disjoint 32-bank groups (conflict-free b64 reads).
// ---------------------------------------------------------------------------
template <bool GELU_OUT, bool RESID, bool ROWSCALE>
__global__ __launch_bounds__(256) void gemm_d128(
    const float* __restrict__ A, const float* __restrict__ W,
    const float* __restrict__ bias, const float* __restrict__ deg,
    float* __restrict__ C, int M)
{
    constexpr int S2 = 144;                   // float2 stride per K-pair row
    __shared__ float sLin[DIM * DIM];         // 64 KB TDM landing buffer
    __shared__ v2f   sWp[64 * S2];            // 73.7 KB paired layout
    __shared__ float sB[DIM];

    const int tid = threadIdx.x;

#if __has_builtin(__builtin_amdgcn_tensor_load_to_lds)
    // ---- Tensor Data Mover: W[128][128] f32 -> sLin (issued by wave 0) ----
    if (tid < 32) {
        const uint64_t ga  = (uint64_t)(uintptr_t)W;
        const uint32_t lds = (uint32_t)(uintptr_t)&sLin[0];   // LDS byte offset
        // D# group0: count=1 | lds_addr | global_addr[56:0] | type=2
        u32x4 g0 = { 1u, lds, (uint32_t)ga,
                     (uint32_t)((ga >> 32) & 0x01FFFFFFu) | 0x80000000u };
        // D# group1: data_size=4B; tensor_dim0/1=128; tile_dim0/1=128;
        //            tensor_dim0_stride=128 (elements); rest 0
        i32x8 g1 = { (int)(2u << 16),          // data_size=2 (4B)
                     (int)(128u << 16),        // tensor_dim0[15:0]=128
                     (int)(128u << 16),        // tensor_dim1[15:0]=128
                     (int)(128u << 16),        // tile_dim0=128
                     128,                      // tile_dim1=128, tile_dim2=0
                     128,                      // tensor_dim0_stride=128
                     0, 0 };
        i32x4 gz4 = { 0, 0, 0, 0 };            // groups 2/3: tile_dim3/4 unused
        i32x8 gz8 = { 0, 0, 0, 0, 0, 0, 0, 0 };
        __builtin_amdgcn_tensor_load_to_lds(g0, g1, gz4, gz4, gz8, 0);
        __builtin_amdgcn_s_wait_tensorcnt(0);
    }
#else
    {
        const float4* __restrict__ W4 = (const float4*)W;
        for (int i = tid; i < DIM * DIM / 4; i += 256)
            *(float4*)(&sLin[i * 4]) = W4[i];
    }
#endif
    if (tid < DIM) sB[tid] = bias[tid];
    __syncthreads();

    // ---- repack linear W into K-paired layout ----
    for (int i = tid; i < DIM * DIM / 4; i += 256) {
        float4 v = *(const float4*)(&sLin[i * 4]);
        const int k = i >> 5;                 // W row
        const int n = (i & 31) << 2;          // W col
        v2f* dst = &sWp[(k >> 1) * S2 + n];
        const int c = k & 1;
        dst[0][c] = v.x; dst[1][c] = v.y; dst[2][c] = v.z; dst[3][c] = v.w;
    }
    __syncthreads();

    const int wave    = tid >> 5;
    const int lane    = tid & 31;
    const int rowTile = blockIdx.x * 8 + wave;
    const int row0    = rowTile * 16;
    if (row0 >= M) return;

    const int half = lane >> 4;      // 0: K lanes {0,1}; 1: K lanes {2,3}
    const int l15  = lane & 15;      // A: row-in-tile; B/C: col-in-tile

    const float* __restrict__ Arow = A + (size_t)(row0 + l15) * DIM;

    float scale = 1.0f;
    if (ROWSCALE) scale = 1.0f / fmaxf(deg[row0 + l15], 1.0f);

    v8f acc[8];
#pragma unroll
    for (int t = 0; t < 8; ++t)
        acc[t] = v8f{0.f, 0.f, 0.f, 0.f, 0.f, 0.f, 0.f, 0.f};

    for (int kk = 0; kk < DIM / 4; ++kk) {
        // A fragment 16x4 f32: lane holds (row=l15, K=kk*4+half*2+{0,1})
        v2f a = *(const v2f*)(Arow + kk * 4 + half * 2);
        if (ROWSCALE) { a.x *= scale; a.y *= scale; }
        const v2f* __restrict__ brow = &sWp[(kk * 2 + half) * S2 + l15];
#pragma unroll
        for (int nt = 0; nt < 8; ++nt) {
            const v2f b = brow[nt * 16];      // one ds_load_b64, aligned pair
            acc[nt] = __builtin_amdgcn_wmma_f32_16x16x4_f32(
                false, a, false, b, (short)0, acc[nt], false, false);
        }
    }

    // C/D layout: VGPR r -> M = r + 8*half, N = l15 (per tile)
    const int mbase = row0 + half * 8;
#pragma unroll
    for (int nt = 0; nt < 8; ++nt) {
        const int col = nt * 16 + l15;
        const float bv = sB[col];
#pragma unroll
        for (int r = 0; r < 8; ++r) {
            float v = acc[nt][r] + bv;
            if (GELU_OUT) v = gelu_f(v);
            if (RESID) v += C[(size_t)(mbase + r) * DIM + col];
            C[(size_t)(mbase + r) * DIM + col] = v;
        }
    }
}

// ---------------------------------------------------------------------------
// Edge phase: one wave32 per edge, 4 floats per lane.
// msg = gelu(key[src] + qry[dst]); agg[dst] += msg  (float atomics -> L2)
// ---------------------------------------------------------------------------
__global__ __launch_bounds__(256) void edge_kernel(
    const float* __restrict__ key, const float* __restrict__ qry,
    const int* __restrict__ src, const int* __restrict__ dst,
    float* __restrict__ agg, int E)
{
    const int e    = (blockIdx.x * blockDim.x + threadIdx.x) >> 5;
    const int lane = threadIdx.x & 31;
    if (e >= E) return;
    const int s = src[e];
    const int d = dst[e];
    const float4 k4 = *(const float4*)(key + (size_t)s * DIM + lane * 4);
    const float4 q4 = *(const float4*)(qry + (size_t)d * DIM + lane * 4);
    float* ap = agg + (size_t)d * DIM + lane * 4;
    atomic_add_f32(ap + 0, gelu_f(k4.x + q4.x));
    atomic_add_f32(ap + 1, gelu_f(k4.y + q4.y));
    atomic_add_f32(ap + 2, gelu_f(k4.z + q4.z));
    atomic_add_f32(ap + 3, gelu_f(k4.w + q4.w));
}

// ---------------------------------------------------------------------------
// out[n] = dot(h[n,:], Wout) + bout   (one wave32 per node, shuffle reduce)
// ---------------------------------------------------------------------------
__global__ __launch_bounds__(256) void out_kernel(
    const float* __restrict__ h, const float* __restrict__ Wout,
    const float* __restrict__ bout, float* __restrict__ out, int N)
{
    const int node = (blockIdx.x * blockDim.x + threadIdx.x) >> 5;
    const int lane = threadIdx.x & 31;
    if (node >= N) return;
    const float4 hv = *(const float4*)(h + (size_t)node * DIM + lane * 4);
    const float4 wv = *(const float4*)(Wout + lane * 4);
    float s = hv.x * wv.x + hv.y * wv.y + hv.z * wv.z + hv.w * wv.w;
#pragma unroll
    for (int off = 16; off > 0; off >>= 1)
        s += __shfl_down(s, off, 32);
    if (lane == 0) out[node] = s + bout[0];
}

// ---------------------------------------------------------------------------
// launch
// ---------------------------------------------------------------------------
extern "C" void kernel_launch(void* const* d_in, const int* in_sizes, int n_in,
                              void* d_out, int out_size, void* d_ws, size_t ws_size,
                              hipStream_t stream)
{
    const float* feats = (const float*)d_in[0];
    const int*   esrc  = (const int*)  d_in[1];
    const int*   edst  = (const int*)  d_in[2];
    const float* Win   = (const float*)d_in[3];
    const float* bin   = (const float*)d_in[4];
    const float* Wk    = (const float*)d_in[5];
    const float* bk    = (const float*)d_in[6];
    const float* Wq    = (const float*)d_in[7];
    const float* bq    = (const float*)d_in[8];
    const float* Wv    = (const float*)d_in[9];
    const float* bv    = (const float*)d_in[10];
    const float* Wl    = (const float*)d_in[11];
    const float* bl    = (const float*)d_in[12];
    const float* Wout  = (const float*)d_in[13];
    const float* bout  = (const float*)d_in[14];
    float* out = (float*)d_out;

    const int N = in_sizes[0] / DIM;   // 50000
    const int E = in_sizes[1];         // 600000
    const size_t ND = (size_t)N * DIM;

    float* h   = (float*)d_ws;
    float* key = h   + ND;
    float* qry = key + ND;
    float* agg = qry + ND;
    float* tmp = agg + ND;
    float* deg = tmp + ND;

    const int mtiles = (N + 15) / 16;
    const dim3 gemm_grid((mtiles + 7) / 8);
    const dim3 blk(256);

    // degree (recomputed every call; ws is not persistent)
    zero_f32<<<(N + 255) / 256, blk, 0, stream>>>(deg, N);
    deg_kernel<<<(E + 255) / 256, blk, 0, stream>>>(edst, deg, E);

    // h = gelu(feats @ Win + bin)
    gemm_d128<true, false, false><<<gemm_grid, blk, 0, stream>>>(
        feats, Win, bin, nullptr, h, N);

    for (int l = 0; l < 3; ++l) {
        const float* Wk_l = Wk + (size_t)l * DIM * DIM;
        const float* Wq_l = Wq + (size_t)l * DIM * DIM;
        const float* Wv_l = Wv + (size_t)l * DIM * DIM;
        const float* Wl_l = Wl + (size_t)l * DIM * DIM;
        const float* bk_l = bk + (size_t)l * DIM;
        const float* bq_l = bq + (size_t)l * DIM;
        const float* bv_l = bv + (size_t)l * DIM;
        const float* bl_l = bl + (size_t)l * DIM;

        // key = h @ Wk + bk ; qry = h @ Wq + bq
        gemm_d128<false, false, false><<<gemm_grid, blk, 0, stream>>>(
            h, Wk_l, bk_l, nullptr, key, N);
        gemm_d128<false, false, false><<<gemm_grid, blk, 0, stream>>>(
            h, Wq_l, bq_l, nullptr, qry, N);

        // agg = segment_sum(gelu(key[src] + qry[dst]), dst)
        zero_f32<<<(int)((ND + 255) / 256), blk, 0, stream>>>(agg, (int)ND);
        edge_kernel<<<(int)(((size_t)E * 32 + 255) / 256), blk, 0, stream>>>(
            key, qry, esrc, edst, agg, E);

        // tmp = gelu((agg / max(deg,1)) @ Wv + bv)
        gemm_d128<true, false, true><<<gemm_grid, blk, 0, stream>>>(
            agg, Wv_l, bv_l, deg, tmp, N);

        // h = tmp @ Wl + bl + h   (residual fused, in-place)
        gemm_d128<false, true, false><<<gemm_grid, blk, 0, stream>>>(
            tmp, Wl_l, bl_l, nullptr, h, N);
    }

    // out = h @ Wout + bout
    out_kernel<<<(int)(((size_t)N * 32 + 255) / 256), blk, 0, stream>>>(
        h, Wout, bout, out, N);
}